// PDASimpleStruct_47296179864288
// MI455X (gfx1250) — compile-verified
//
#include <hip/hip_runtime.h>

// Problem constants (from the reference)
#define T_STEPS 512
#define BATCH   128
#define EMB     128
#define SSLOTS  1024          // 2*T
#define NWG     4             // phase-1 workgroups (4 x 32 batches = 128)
#define BPG     (BATCH/NWG)   // batches per workgroup (one wave per batch)

typedef float v2f __attribute__((ext_vector_type(2)));
typedef float v8f __attribute__((ext_vector_type(8)));

// ---- ordered-uint encoding so unsigned atomicMax == float max ------------
__device__ __forceinline__ unsigned f32_to_key(float f) {
  unsigned b = __float_as_uint(f);
  return (b & 0x80000000u) ? ~b : (b | 0x80000000u);
}
__device__ __forceinline__ float key_to_f32(unsigned k) {
  unsigned b = (k & 0x80000000u) ? (k ^ 0x80000000u) : ~k;
  return __uint_as_float(b);
}

// ---- init: zero the cross-batch max buffer M and barrier counters --------
__global__ void pda_init_kernel(unsigned* __restrict__ M, int* __restrict__ gcnt) {
  int i = blockIdx.x * blockDim.x + threadIdx.x;
  if (i < T_STEPS * SSLOTS) M[i] = 0u;   // 0 is below every encoded float
  if (i < T_STEPS)          gcnt[i] = 0;
}

// ---- phase 1: sequential scan producing W[b][t][s] -----------------------
// 4 workgroups x 1024 threads. Wave w <-> batch (blockIdx.x*32 + w).
// Lane l owns slots {32c + l} of its batch (interleaved -> conflict-free LDS).
__global__ __launch_bounds__(1024, 1)
void pda_scan_kernel(const float* __restrict__ d1, const float* __restrict__ d2,
                     const float* __restrict__ u,
                     float* __restrict__ W, unsigned* __restrict__ M,
                     int* __restrict__ gcnt)
{
  __shared__ float    sstr[BPG][SSLOTS];   // 128 KB of the 320 KB WGP LDS
  __shared__ unsigned scmax[SSLOTS];       // per-WG cross-batch max (encoded)

  const int tid  = threadIdx.x;
  const int wave = tid >> 5;
  const int lane = tid & 31;
  const int b    = blockIdx.x * BPG + wave;

  for (int i = tid; i < BPG * SSLOTS; i += 1024) (&sstr[0][0])[i] = 0.f;
  __syncthreads();

  float sval[32];   // this lane's 32 strengths (pre-pop) for the current step
  float cand[32];   // u_b - used  per owned slot

  for (int t = 0; t < T_STEPS; ++t) {
    scmax[tid] = 0u;                                // reset per-step max
    if (lane == 0) {                                // push two entries
      sstr[wave][2 * t]     = d1[t * BATCH + b];
      sstr[wave][2 * t + 1] = d2[t * BATCH + b];
    }
    __syncthreads();

    const float ub = u[t * BATCH + b];

    // exclusive reverse cumsum: chunks 31 -> 0, wave suffix-scan per chunk
    float carry = 0.f;
    #pragma unroll
    for (int c = 31; c >= 0; --c) {
      float v = sstr[wave][c * 32 + lane];
      sval[c] = v;
      float s = v;
      #pragma unroll
      for (int off = 1; off < 32; off <<= 1) {      // inclusive suffix scan
        float o = __shfl_down(s, off);
        if (lane + off < 32) s += o;
      }
      float used = (s - v) + carry;                 // strength above this slot
      cand[c] = ub - used;
      carry += __shfl(s, 0);                        // chunk total
      atomicMax(&scmax[c * 32 + lane], f32_to_key(cand[c]));  // ds_max_u32
    }
    __syncthreads();

    // one GLOBAL_ATOMIC_MAX_U32 per thread into this step's row of M
    atomicMax(&M[t * SSLOTS + tid], scmax[tid]);

    // 4-way inter-workgroup barrier for step t
    __threadfence();
    if (tid == 0) {
      atomicAdd(&gcnt[t], 1);
      while (__hip_atomic_load(&gcnt[t], __ATOMIC_ACQUIRE,
                               __HIP_MEMORY_SCOPE_AGENT) < NWG)
        __builtin_amdgcn_s_sleep(1);
    }
    __syncthreads();
    __threadfence();

    // read m, emit w = min(strength, m), pop
    const int cstore = min(31, ((2 * t + 1) >> 5) + 1);  // covers s <= 2t+31
    #pragma unroll
    for (int c = 0; c < 32; ++c) {
      float m = key_to_f32(M[t * SSLOTS + c * 32 + lane]);
      float w = fminf(sval[c], m);                  // may be negative: faithful
      if (c <= cstore)
        W[(((size_t)b) * T_STEPS + t) * SSLOTS + c * 32 + lane] = w;
      float pop = fminf(sval[c], fmaxf(0.f, cand[c]));
      sstr[wave][c * 32 + lane] = sval[c] - pop;
    }
    __syncthreads();
  }
}

// ---- phase 2: per-batch GEMM out_b[T,E] = W_b[T,S] @ V_b[S,E] via WMMA ---
// V_b[s,:] = (s even ? v1 : v2)[s/2, b, :].  8 waves/block, one 16x16 D-tile
// per wave (16 t-rows x 16 e-cols), K-loop in steps of 4 (f32 WMMA shape).
__global__ __launch_bounds__(256, 4)
void pda_gemm_kernel(const float* __restrict__ v1, const float* __restrict__ v2,
                     const float* __restrict__ W, float* __restrict__ out)
{
  const int lane = threadIdx.x & 31;
  const int wave = threadIdx.x >> 5;                // 0..7
  const int e0 = blockIdx.x * 16;
  const int t0 = blockIdx.y * 128 + wave * 16;
  const int b  = blockIdx.z;

  const int row = lane & 15;                        // A: M index
  const int hi  = lane >> 4;                        // 0: K=0,1  1: K=2,3
  const int col = lane & 15;                        // B/D: N index

  const float* wrow = W + (((size_t)b) * T_STEPS + (t0 + row)) * SSLOTS + hi * 2;
  const float* v1p  = v1 + (size_t)b * EMB + e0 + col;
  const float* v2p  = v2 + (size_t)b * EMB + e0 + col;

  v8f acc = {0.f, 0.f, 0.f, 0.f, 0.f, 0.f, 0.f, 0.f};

  // w[t,s] == 0 for s > 2t+1 -> only K up to 2*t0+31 matters for this tile
  const int kend = (2 * t0 + 32) >> 2;              // <= 256
  #pragma unroll 4
  for (int k = 0; k < kend; ++k) {
    const int s0 = k * 4;
    v2f a = *(const v2f*)(wrow + s0);               // A[16x4] tile, K=hi*2..+1
    const int tp = (s0 >> 1) + hi;                  // push step for K rows
    v2f bm;
    bm.x = v1p[(size_t)tp * (BATCH * EMB)];         // B[K even, col]
    bm.y = v2p[(size_t)tp * (BATCH * EMB)];         // B[K odd , col]
    acc = __builtin_amdgcn_wmma_f32_16x16x4_f32(
        /*neg_a=*/false, a, /*neg_b=*/false, bm,
        /*c_mod=*/(short)0, acc, /*reuse_a=*/false, /*reuse_b=*/false);
  }

  // D layout: VGPR r -> M = r + hi*8 ; lanes give N = col
  float* orow = out + (size_t)(t0 + hi * 8) * (BATCH * EMB)
                    + (size_t)b * EMB + e0 + col;
  #pragma unroll
  for (int r = 0; r < 8; ++r)
    orow[(size_t)r * (BATCH * EMB)] = acc[r];
}

extern "C" void kernel_launch(void* const* d_in, const int* in_sizes, int n_in,
                              void* d_out, int out_size, void* d_ws, size_t ws_size,
                              hipStream_t stream) {
  const float* v1 = (const float*)d_in[0];
  const float* v2 = (const float*)d_in[1];
  const float* d1 = (const float*)d_in[2];
  const float* d2 = (const float*)d_in[3];
  const float* u  = (const float*)d_in[4];
  float* out = (float*)d_out;

  // workspace layout (needs ~271 MB): W | M | gcnt
  char* ws = (char*)d_ws;
  const size_t W_bytes = (size_t)BATCH * T_STEPS * SSLOTS * sizeof(float);
  const size_t M_bytes = (size_t)T_STEPS * SSLOTS * sizeof(unsigned);
  float*    W    = (float*)ws;
  unsigned* M    = (unsigned*)(ws + W_bytes);
  int*      gcnt = (int*)(ws + W_bytes + M_bytes);

  pda_init_kernel<<<(T_STEPS * SSLOTS + 255) / 256, 256, 0, stream>>>(M, gcnt);
  pda_scan_kernel<<<NWG, 1024, 0, stream>>>(d1, d2, u, W, M, gcnt);
  dim3 g2(EMB / 16, T_STEPS / 128, BATCH);
  pda_gemm_kernel<<<g2, 256, 0, stream>>>(v1, v2, W, out);
}